// GraphConvSparse_2241972928585
// MI455X (gfx1250) — compile-verified
//
#include <hip/hip_runtime.h>

// GraphConvSparse: out[col[e], :] += nan_to_num(w[e]) * x[row[e], :]
// N=100000 nodes, H=64 features, E=1.25M edges.
//
// Atomic-scatter formulation: x (25.6MB) and out (25.6MB) are both L2-resident
// (192MB L2), so the random gather + atomic scatter stay on-chip after first
// touch. Main kernel: each wave32 handles EPW=8 edges with NO per-edge guards
// (tail edges go to a separate tiny kernel); edge scalars load as merged wide
// SMEM clauses (s_load_b512 x2 + s_load_b256). All gather/scatter addresses
// are 32-bit element offsets (row*H + 2*lane fits i32) so loads/atomics keep
// saddr-form: SGPR base pair + one 32-bit VGPR offset per edge.

#define H   64
#define EPW 8   // edges per wave in the main kernel

// ---------------------------------------------------------------------------
// Main scatter kernel: full EPW-edge waves only, zero per-edge guards.
// ---------------------------------------------------------------------------
__global__ __launch_bounds__(256) void gcs_scatter_main(
    const float*     __restrict__ x,     // [N, H]
    const long long* __restrict__ ei64,  // [2*E] int64 edge_index (row | col)
    const float*     __restrict__ w,     // [E]
    float*           __restrict__ out,   // [N, H]
    int E,                               // total edges (for col base offset)
    int nWaves) {                        // number of full EPW-edge waves
    const int gtid = blockIdx.x * blockDim.x + threadIdx.x;
    const int lane = threadIdx.x & 31;

    // Wave-uniform edge base; readfirstlane makes uniformity provable so all
    // per-edge scalars ride SMEM and addresses stay saddr-form.
    const int wid = __builtin_amdgcn_readfirstlane(gtid >> 5);
    if (wid >= nWaves) return;           // whole-wave scalar guard only
    const int ebase = wid * EPW;

    // --- Edge scalars: contiguous -> merged wide SMEM loads ------------------
    int   row[EPW];
    int   col[EPW];
    float wt[EPW];
#pragma unroll
    for (int k = 0; k < EPW; ++k) {
        row[k] = (int)ei64[ebase + k];        // 8 contig int64 -> s_load_b512
        col[k] = (int)ei64[E + ebase + k];    // 8 contig int64 -> s_load_b512
        float t = w[ebase + k];               // 8 contig f32   -> s_load_b256
        // reference: jnp.nan_to_num(w, nan=0, posinf=0, neginf=0)
        wt[k] = __builtin_isfinite(t) ? t : 0.0f;
    }

    const int foff = 2 * lane;           // feature offset (floats), lane-local

    // --- Gather all rows first (8 outstanding loads overlap L2 latency) ------
    // 32-bit element offsets: row*H + foff < N*H = 6.4M, fits i32 -> GVS
    // addressing (SGPR base + 32-bit VGPR offset), no 64-bit address VGPRs.
    float2 v[EPW];
#pragma unroll
    for (int k = 0; k < EPW; ++k) {
        const int off = row[k] * H + foff;
        v[k] = *reinterpret_cast<const float2*>(x + off);
    }

    // --- Scatter: non-returning hardware FP32 atomics into L2 ----------------
#pragma unroll
    for (int k = 0; k < EPW; ++k) {
        const int off = col[k] * H + foff;
        float* dst = out + off;
        __hip_atomic_fetch_add(dst + 0, v[k].x * wt[k], __ATOMIC_RELAXED,
                               __HIP_MEMORY_SCOPE_AGENT);
        __hip_atomic_fetch_add(dst + 1, v[k].y * wt[k], __ATOMIC_RELAXED,
                               __HIP_MEMORY_SCOPE_AGENT);
    }
}

// ---------------------------------------------------------------------------
// Tail kernel: edges [start, E), one edge per wave, single block.
// (rem < EPW, so one 256-thread block = 8 waves always suffices.)
// ---------------------------------------------------------------------------
__global__ __launch_bounds__(256) void gcs_scatter_tail(
    const float*     __restrict__ x,
    const long long* __restrict__ ei64,
    const float*     __restrict__ w,
    float*           __restrict__ out,
    int E, int start) {
    const int lane = threadIdx.x & 31;
    const int wid  = __builtin_amdgcn_readfirstlane((int)threadIdx.x >> 5);
    const int e    = start + wid;
    if (e >= E) return;

    const int row = (int)ei64[e];
    const int col = (int)ei64[E + e];
    float t = w[e];
    const float wt = __builtin_isfinite(t) ? t : 0.0f;

    const int foff = 2 * lane;
    const float2 v = *reinterpret_cast<const float2*>(x + row * H + foff);
    float* dst = out + col * H + foff;
    __hip_atomic_fetch_add(dst + 0, v.x * wt, __ATOMIC_RELAXED,
                           __HIP_MEMORY_SCOPE_AGENT);
    __hip_atomic_fetch_add(dst + 1, v.y * wt, __ATOMIC_RELAXED,
                           __HIP_MEMORY_SCOPE_AGENT);
}

// ---------------------------------------------------------------------------
// Zero the output (required every call: atomics accumulate and the harness
// replays the captured graph without re-poisoning d_out).
// ---------------------------------------------------------------------------
__global__ __launch_bounds__(256) void gcs_zero_kernel(float4* __restrict__ out,
                                                       int n4) {
    int i = blockIdx.x * blockDim.x + threadIdx.x;
    if (i < n4) {
        out[i] = make_float4(0.f, 0.f, 0.f, 0.f);
    }
}

// ---------------------------------------------------------------------------
// Launch
//   d_in[0] = inputs      float32 [N*H]
//   d_in[1] = edge_index  int64   [2*E]
//   d_in[2] = edge_weight float32 [E]
//   d_out   = output      float32 [N*H]
// ---------------------------------------------------------------------------
extern "C" void kernel_launch(void* const* d_in, const int* in_sizes, int n_in,
                              void* d_out, int out_size, void* d_ws, size_t ws_size,
                              hipStream_t stream) {
    const float*     x    = (const float*)d_in[0];
    const long long* ei64 = (const long long*)d_in[1];
    const float*     w    = (const float*)d_in[2];
    float*           out  = (float*)d_out;

    const int E = in_sizes[2];                 // 1,250,000

    // Zero output (float4 fill).
    const int n4 = out_size / 4;               // N*H/4
    {
        dim3 blk(256);
        dim3 grd((n4 + 255) / 256);
        gcs_zero_kernel<<<grd, blk, 0, stream>>>((float4*)out, n4);
    }

    // Main scatter: EPW edges per wave32, 8 waves per 256-thread block.
    const int nWaves = E / EPW;                // full waves
    const int nFull  = nWaves * EPW;           // edges covered by main kernel
    if (nWaves > 0) {
        dim3 blk(256);
        const int waves_per_block = 256 / 32;
        dim3 grd((nWaves + waves_per_block - 1) / waves_per_block);
        gcs_scatter_main<<<grd, blk, 0, stream>>>(x, ei64, w, out, E, nWaves);
    }

    // Tail scatter: remaining E % EPW edges (one block, one edge per wave).
    if (nFull < E) {
        gcs_scatter_tail<<<1, 256, 0, stream>>>(x, ei64, w, out, E, nFull);
    }
}